// AvgPooling_39238821216544
// MI455X (gfx1250) — compile-verified
//
#include <hip/hip_runtime.h>
#include <math.h>

typedef __attribute__((ext_vector_type(2))) float v2f;
typedef __attribute__((ext_vector_type(8))) float v8f;

#define B_TOT 16384
#define LSEQ  50
#define DIM   64
#define NATTR 16
#define USERS_PER_BLOCK 16   // 16 waves (wave32) per block, one user per wave

// ---------------------------------------------------------------------------
// Fused: mean-pool embeddings -> LDS -> WMMA projection (16x16, K=64 via 16
// chained v_wmma_f32_16x16x4_f32) -> per-group softmax / log-softmax ->
// logits to d_out + per-block (num,cnt) loss partials to workspace.
// Grid: B/16 = 1024 blocks x 512 threads.
// ---------------------------------------------------------------------------
__global__ __launch_bounds__(512) void fused_pool_proj_softmax(
    const int*   __restrict__ x,
    const int*   __restrict__ x_mask,
    const float* __restrict__ y,
    const float* __restrict__ ob,
    const float* __restrict__ item_emb,
    const float* __restrict__ w0, const float* __restrict__ b0,
    const float* __restrict__ w1, const float* __restrict__ b1,
    const float* __restrict__ w2, const float* __restrict__ b2,
    float* __restrict__ logit_out,
    float* __restrict__ partials)   // [gridDim.x][6] = num0..2, cnt0..2
{
    __shared__ float rep[USERS_PER_BLOCK][DIM + 4];  // stride 68: conflict-free column reads
    __shared__ float Wu[USERS_PER_BLOCK][NATTR + 1]; // projected logits (pre-softmax)

    const int tid  = threadIdx.x;
    const int wid  = tid >> 5;    // wave id == local user id
    const int lane = tid & 31;
    const int user = blockIdx.x * USERS_PER_BLOCK + wid;

    // ---------------- mean pooling (one wave per user) ----------------
    // x==0 rows map to item_emb[0]==0, so summing all L positions is exact.
    const int* xrow = x + user * LSEQ;
    const int  xi_a = xrow[lane];                                    // positions 0..31
    const int  xi_b = (lane < LSEQ - 32) ? xrow[32 + lane] : 0;      // positions 32..49

    int msum = x_mask[user * LSEQ + lane]
             + ((lane < LSEQ - 32) ? x_mask[user * LSEQ + 32 + lane] : 0);
#pragma unroll
    for (int off = 16; off >= 1; off >>= 1) msum += __shfl_xor(msum, off, 32);
    const float inv_len = 1.0f / (float)msum;                        // x_len >= 1 by construction

    float s0 = 0.0f, s1 = 0.0f;                                      // lane owns dims 2*lane, 2*lane+1
#pragma unroll
    for (int l = 0; l < LSEQ; ++l) {
        const int idx = (l < 32) ? __shfl(xi_a, l, 32) : __shfl(xi_b, l - 32, 32);
        const float2 e = *(const float2*)(item_emb + (size_t)idx * DIM + 2 * lane);
        s0 += e.x;
        s1 += e.y;
    }
    rep[wid][2 * lane]     = s0 * inv_len;
    rep[wid][2 * lane + 1] = s1 * inv_len;
    __syncthreads();

    // ---------------- WMMA projection: D(16x16) = rep(16x64) @ W^T(64x16) ----------------
    // Executed by wave 0 only (branch is wave-uniform -> EXEC all-ones inside).
    if (wid == 0) {
        const int half = lane >> 4;     // 0: K even pair base, 1: +2
        const int mn   = lane & 15;     // A row M (for SRC0) == B col N (for SRC1)

        // Row mn of concatenated W = [w0; w1; w2] (16 x 64), and its bias.
        const float* wr;
        float bias;
        if      (mn < 2) { wr = w0 + mn * DIM;       bias = b0[mn];     }
        else if (mn < 8) { wr = w1 + (mn - 2) * DIM; bias = b1[mn - 2]; }
        else             { wr = w2 + (mn - 8) * DIM; bias = b2[mn - 8]; }

        v8f acc = {0.f, 0.f, 0.f, 0.f, 0.f, 0.f, 0.f, 0.f};
#pragma unroll
        for (int s = 0; s < 16; ++s) {
            const int k = 4 * s + 2 * half;
            v2f a, b;
            a.x = rep[mn][k];     // A: lanes 0-15 hold K=k0,k0+1; lanes 16-31 K=k0+2,k0+3
            a.y = rep[mn][k + 1];
            b.x = wr[k];          // B[k][n] = W[n][k]
            b.y = wr[k + 1];
            acc = __builtin_amdgcn_wmma_f32_16x16x4_f32(
                      false, a, false, b, (short)0, acc, false, false);
        }
        // C/D layout: VGPR r -> (M = r + 8*half, N = mn)
#pragma unroll
        for (int r = 0; r < 8; ++r)
            Wu[r + 8 * half][mn] = acc[r] + bias;
    }
    __syncthreads();

    // ---------------- per-group log-softmax, loss partials, logits ----------------
    float num0 = 0.f, num1 = 0.f, num2 = 0.f;
    float cnt0 = 0.f, cnt1 = 0.f, cnt2 = 0.f;
    if (wid == 0) {
        if (lane < 16) {
            const int m = lane;
            const int gu = blockIdx.x * USERS_PER_BLOCK + m;
            float v[NATTR];
#pragma unroll
            for (int j = 0; j < NATTR; ++j) v[j] = Wu[m][j];

            const float* yrow  = y  + (size_t)gu * NATTR;
            const float* obrow = ob + (size_t)gu * NATTR;
            float*       lrow  = logit_out + (size_t)gu * NATTR;

            const int gs[4] = {0, 2, 8, 16};
            float nums[3], cnts[3];
#pragma unroll
            for (int g = 0; g < 3; ++g) {
                const int s = gs[g], e = gs[g + 1];
                float mx = -3.402823466e+38f;
                for (int j = s; j < e; ++j) mx = fmaxf(mx, v[j]);
                float se = 0.f;
                for (int j = s; j < e; ++j) se += expf(v[j] - mx);
                const float lse = mx + logf(se);
                const float inv = 1.0f / se;
                const float obs = obrow[s];   // ob columns within a group are identical
                float accl = 0.f;
                for (int j = s; j < e; ++j) {
                    const float lp = v[j] - lse;
                    accl  += yrow[j] * lp;
                    lrow[j] = expf(v[j] - mx) * inv;   // softmax
                }
                nums[g] = -accl * obs;
                cnts[g] = obs;
            }
            num0 = nums[0]; num1 = nums[1]; num2 = nums[2];
            cnt0 = cnts[0]; cnt1 = cnts[1]; cnt2 = cnts[2];
        }
        // full-wave reduction (lanes 16-31 contribute zeros)
#pragma unroll
        for (int off = 16; off >= 1; off >>= 1) {
            num0 += __shfl_xor(num0, off, 32);
            num1 += __shfl_xor(num1, off, 32);
            num2 += __shfl_xor(num2, off, 32);
            cnt0 += __shfl_xor(cnt0, off, 32);
            cnt1 += __shfl_xor(cnt1, off, 32);
            cnt2 += __shfl_xor(cnt2, off, 32);
        }
        if (lane == 0) {
            float* p = partials + (size_t)blockIdx.x * 6;
            p[0] = num0; p[1] = num1; p[2] = num2;
            p[3] = cnt0; p[4] = cnt1; p[5] = cnt2;
        }
    }
}

// ---------------------------------------------------------------------------
// Deterministic final reduction: sum 1024 x 6 partials, compute
// loss = sum_g num_g / max(cnt_g, 1)
// ---------------------------------------------------------------------------
__global__ __launch_bounds__(256) void finalize_loss(
    const float* __restrict__ partials, int nblk, float* __restrict__ loss_out)
{
    __shared__ float red[256];
    const int t = threadIdx.x;
    float tot[6];
    for (int g = 0; g < 6; ++g) {
        float s = 0.f;
        for (int i = t; i < nblk; i += 256) s += partials[(size_t)i * 6 + g];
        red[t] = s;
        __syncthreads();
        for (int stride = 128; stride > 0; stride >>= 1) {
            if (t < stride) red[t] += red[t + stride];
            __syncthreads();
        }
        tot[g] = red[0];
        __syncthreads();
    }
    if (t == 0) {
        float loss = 0.f;
        for (int g = 0; g < 3; ++g) loss += tot[g] / fmaxf(tot[3 + g], 1.0f);
        loss_out[0] = loss;
    }
}

extern "C" void kernel_launch(void* const* d_in, const int* in_sizes, int n_in,
                              void* d_out, int out_size, void* d_ws, size_t ws_size,
                              hipStream_t stream) {
    const int*   x    = (const int*)d_in[0];
    const int*   xm   = (const int*)d_in[1];
    const float* y    = (const float*)d_in[2];
    const float* ob   = (const float*)d_in[3];
    const float* emb  = (const float*)d_in[4];
    const float* w0   = (const float*)d_in[5];
    const float* b0   = (const float*)d_in[6];
    const float* w1   = (const float*)d_in[7];
    const float* b1   = (const float*)d_in[8];
    const float* w2   = (const float*)d_in[9];
    const float* b2   = (const float*)d_in[10];

    float* out      = (float*)d_out;
    float* partials = (float*)d_ws;          // 1024 * 6 floats = 24 KB

    const int nblk = B_TOT / USERS_PER_BLOCK;  // 1024
    fused_pool_proj_softmax<<<nblk, 512, 0, stream>>>(
        x, xm, y, ob, emb, w0, b0, w1, b1, w2, b2, out, partials);
    finalize_loss<<<1, 256, 0, stream>>>(partials, nblk, out + (size_t)B_TOT * NATTR);
}